// SegmentPool_71683004171095
// MI455X (gfx1250) — compile-verified
//
#include <hip/hip_runtime.h>
#include <cstdint>
#include <cstddef>

// ---------------------------------------------------------------------------
// segment_sum(x[N,64] f32, idx[N] i32, 100k segments) -> out[100k,64] f32
// Memory-bound (0.25 FLOP/byte): stream x via TDM into LDS (double-buffered),
// scatter with hardware global_atomic_add_f32 (output is L2-resident: 25.6MB
// against 192MB L2). No WMMA: matrix path would inflate traffic ~1500x.
// ---------------------------------------------------------------------------

typedef unsigned int v4u __attribute__((ext_vector_type(4)));
typedef int          v8i __attribute__((ext_vector_type(8)));
typedef int          v4i __attribute__((ext_vector_type(4)));

#define D_COLS        64
#define TILE_ROWS     64          // rows staged per block per tile (16 KB)
#define ROWS_PER_WAVE 8           // 8 waves (wave32) x 8 rows = 64 rows
#define BLOCK_THREADS 256

#if defined(__has_builtin)
#if __has_builtin(__builtin_amdgcn_tensor_load_to_lds) && \
    __has_builtin(__builtin_amdgcn_s_wait_tensorcnt)
#define USE_TDM 1
#endif
#endif
#ifndef USE_TDM
#define USE_TDM 0
#endif

// ---------------------------------------------------------------------------
__global__ void __launch_bounds__(BLOCK_THREADS)
zero_f32_kernel(float* __restrict__ out, int n) {
  int tid    = blockIdx.x * blockDim.x + threadIdx.x;
  int stride = gridDim.x * blockDim.x;
  int n4     = n >> 2;
  float4* o4 = reinterpret_cast<float4*>(out);
  float4 z{0.f, 0.f, 0.f, 0.f};
  for (int j = tid; j < n4; j += stride) o4[j] = z;
  for (int j = (n4 << 2) + tid; j < n; j += stride) out[j] = 0.f;  // tail
}

// ---------------------------------------------------------------------------
// Scatter one staged LDS tile into out[] with HW f32 atomics.
// 16 lanes cover one 64-float row as float4s; 4 passes cover 64 rows.
__device__ __forceinline__ void consume_tile(const float4* lb,
                                             const int* __restrict__ idx,
                                             float* __restrict__ out,
                                             int tile, int nRows, int tid) {
  const int r0 = tid >> 4;        // 0..15
  const int c4 = tid & 15;        // float4 column group 0..15
#pragma unroll
  for (int p = 0; p < 4; ++p) {
    const int r   = (p << 4) + r0;
    const int row = tile * TILE_ROWS + r;
    if (row < nRows) {
      const int    seg = idx[row];
      const float4 v   = lb[r * 16 + c4];          // ds_load_b128
      float* o = out + (size_t)seg * D_COLS + (c4 << 2);
      unsafeAtomicAdd(o + 0, v.x);                 // global_atomic_add_f32
      unsafeAtomicAdd(o + 1, v.y);
      unsafeAtomicAdd(o + 2, v.z);
      unsafeAtomicAdd(o + 3, v.w);
    }
  }
}

#if USE_TDM
// Issue one per-wave TDM descriptor: tile of [rows x 64] f32, row-major,
// global -> LDS. D# layout per CDNA5 ISA §8.3-8.6. This toolchain's builtin
// takes 6 operands: (v4u g0, v8i g1, v4i g2, v4i g3, v8i extra, i32 cpol);
// groups 2/3 and the extra descriptor words are zero for a 2-D tensor.
__device__ __forceinline__ void tdm_issue(const float* __restrict__ x,
                                          int nRows, int tile, int bufSel,
                                          unsigned ldsBase, int wv) {
  const int row0 = tile * TILE_ROWS + wv * ROWS_PER_WAVE;
  int rows = nRows - row0;
  if (rows > ROWS_PER_WAVE) rows = ROWS_PER_WAVE;
  if (rows <= 0) return;  // scalar (wave-uniform) skip: no stray DMA

  const unsigned long long gaddr =
      (unsigned long long)(uintptr_t)x +
      (unsigned long long)row0 * (D_COLS * 4ull);
  const unsigned lds = ldsBase +
      (unsigned)bufSel * (TILE_ROWS * D_COLS * 4u) +
      (unsigned)wv * (ROWS_PER_WAVE * D_COLS * 4u);

  // --- D# group 0 (128b): count=1 | lds_addr | global_addr | type=2 ---
  v4u g0;
  g0.x = 1u;                                            // count=1, user mode
  g0.y = lds;                                           // lds_addr (bytes)
  g0.z = (unsigned)(gaddr & 0xFFFFFFFFull);             // global_addr[31:0]
  g0.w = (unsigned)((gaddr >> 32) & 0x1FFFFFFull)       // global_addr[56:32]
       | (2u << 30);                                    // type=2 ("image")
  // --- D# group 1 (256b) ---
  v8i g1;
  g1[0] = (int)(2u << 16);                              // data_size=4B; mask=0
  g1[1] = (int)((unsigned)D_COLS << 16);                // tensor_dim0[15:0]=64
  g1[2] = (int)(((unsigned)rows & 0xFFFFu) << 16);      // tensor_dim1[15:0]
  g1[3] = (int)((unsigned)D_COLS << 16);                // tile_dim0=64
  g1[4] = rows;                                         // tile_dim1; tile_dim2=0
  g1[5] = D_COLS;                                       // tensor_dim0_stride=64
  g1[6] = 0;                                            // dim1_stride (unused 2D)
  g1[7] = 0;
  v4i g2 = {0, 0, 0, 0};                                // no dims 2..4
  v4i g3 = {0, 0, 0, 0};
  v8i g4 = {0, 0, 0, 0, 0, 0, 0, 0};                    // extra words (unused)
  __builtin_amdgcn_tensor_load_to_lds(g0, g1, g2, g3, g4, 0);
}
#endif

// ---------------------------------------------------------------------------
__global__ void __launch_bounds__(BLOCK_THREADS)
segsum_kernel(const float* __restrict__ x, const int* __restrict__ idx,
              float* __restrict__ out, int nRows) {
  __shared__ float4 buf[2][TILE_ROWS * 16];   // 2 x 16 KB double buffer

  const int tid    = threadIdx.x;
  const int nTiles = (nRows + TILE_ROWS - 1) / TILE_ROWS;

#if USE_TDM
  const int wv = __builtin_amdgcn_readfirstlane((int)(tid >> 5));  // wave id
  const unsigned ldsBase = (unsigned)(uintptr_t)(void*)&buf[0][0]; // LDS byte off

  int t = blockIdx.x;
  if (t < nTiles) tdm_issue(x, nRows, t, 0, ldsBase, wv);  // prime pipeline
  int b = 0;
  for (; t < nTiles; t += gridDim.x) {
    const int  tn      = t + (int)gridDim.x;
    const bool hasNext = tn < nTiles;
    if (hasNext) {
      tdm_issue(x, nRows, tn, b ^ 1, ldsBase, wv);   // overlap next DMA
      __builtin_amdgcn_s_wait_tensorcnt(1);          // tile t landed
    } else {
      __builtin_amdgcn_s_wait_tensorcnt(0);
    }
    __syncthreads();                                  // all waves' slices ready
    consume_tile(buf[b], idx, out, t, nRows, tid);
    __syncthreads();                                  // safe to overwrite buf[b]
    b ^= 1;
  }
#else
  // Fallback: stage via regular global loads (still LDS-tiled).
  const float4* x4 = reinterpret_cast<const float4*>(x);
  for (int t = blockIdx.x; t < nTiles; t += gridDim.x) {
    const int base   = t * TILE_ROWS * 16;           // float4 index into x
    const int limit4 = nRows * 16;
    for (int k = tid; k < TILE_ROWS * 16; k += BLOCK_THREADS) {
      const int g = base + k;
      float4 v{0.f, 0.f, 0.f, 0.f};
      if (g < limit4) v = x4[g];
      buf[0][k] = v;
    }
    __syncthreads();
    consume_tile(buf[0], idx, out, t, nRows, tid);
    __syncthreads();
  }
#endif
}

// ---------------------------------------------------------------------------
extern "C" void kernel_launch(void* const* d_in, const int* in_sizes, int n_in,
                              void* d_out, int out_size, void* d_ws, size_t ws_size,
                              hipStream_t stream) {
  const float* x   = reinterpret_cast<const float*>(d_in[0]);
  const int*   idx = reinterpret_cast<const int*>(d_in[1]);
  float*       out = reinterpret_cast<float*>(d_out);
  const int nRows  = in_sizes[1];          // N (idx element count)

  // Pass 1: zero the accumulator (harness poisons d_out; atomics accumulate).
  zero_f32_kernel<<<1024, BLOCK_THREADS, 0, stream>>>(out, out_size);

  // Pass 2: TDM-staged scatter-add.
  const int nTiles = (nRows + TILE_ROWS - 1) / TILE_ROWS;
  int grid = nTiles < 4096 ? (nTiles > 0 ? nTiles : 1) : 4096;
  segsum_kernel<<<grid, BLOCK_THREADS, 0, stream>>>(x, idx, out, nRows);
}